// MMPhysicsFeatureComputer_90941637526146
// MI455X (gfx1250) — compile-verified
//
#include <hip/hip_runtime.h>

// ---------------------------------------------------------------------------
// MM electrostatics features on gfx1250 (wave32, WMMA f32 16x16x4)
//   s_ep[b,i]   = sum_j q_j / (|r_i - r_j| + eps)
//   v_ef[b,i,:] = sum_j q_j (r_i - r_j) / (|r_i - r_j| + eps)^3
// d^2 tiles computed on the matrix unit via:
//   d^2[i][j] = |r_i|^2 (C-init) + A(x_i,y_i,z_i,1) . B(-2x_j,-2y_j,-2z_j,|r_j|^2)
// Field assembled as E_i = r_i * S3_i - T_i  (both linear in per-pair weights)
// so M is split across waves and combined with fp32 global atomics.
// ---------------------------------------------------------------------------

typedef float v2f __attribute__((ext_vector_type(2)));
typedef float v8f __attribute__((ext_vector_type(8)));

#define EPSV 1e-8f

constexpr int Bc = 32;     // batches
constexpr int Nc = 128;    // QM atoms
constexpr int Mc = 4096;   // MM atoms
constexpr int ROW_TILES = Nc / 16;        // 8 row tiles of 16 QM atoms
constexpr int MSPLIT = 8;                 // M split across waves
constexpr int MCHUNK = Mc / MSPLIT;       // 512 MM atoms per wave
constexpr int WAVES_PER_BLOCK = 8;        // 256 threads = 8 wave32

__global__ void zero_kernel(float* __restrict__ out, int n) {
    int i = blockIdx.x * blockDim.x + threadIdx.x;
    if (i < n) out[i] = 0.0f;
}

__global__ __launch_bounds__(32 * WAVES_PER_BLOCK)
void mm_features_kernel(const float* __restrict__ posQ,   // [B,N,3]
                        const float* __restrict__ posM,   // [B,M,3]
                        const float* __restrict__ chg,    // [B,M]
                        const int*   __restrict__ nmm,    // [B]
                        float* __restrict__ out)          // [B*N] ++ [B*N*3]
{
    const int lane = threadIdx.x & 31;
    const int wave = threadIdx.x >> 5;
    const int w    = blockIdx.x * WAVES_PER_BLOCK + wave;

    const int b   = w / (ROW_TILES * MSPLIT);
    const int rem = w % (ROW_TILES * MSPLIT);
    const int rt  = rem / MSPLIT;           // QM row tile
    const int mc  = rem % MSPLIT;           // M chunk

    const int rowBase = rt * 16;
    const int col     = lane & 15;
    const bool hi     = lane >= 16;

    // ---- A operand: QM tile, row = rowBase + col, K = (x, y, z, 1) --------
    const float* qp = posQ + ((size_t)b * Nc + rowBase + col) * 3;
    const float qx = qp[0], qy = qp[1], qz = qp[2];
    const float rn2 = qx * qx + qy * qy + qz * qz;

    v2f a;
    a.x = hi ? qz   : qx;   // lanes 0-15: K0,K1 ; lanes 16-31: K2,K3
    a.y = hi ? 1.0f : qy;

    // ---- C init: c[k] = |r_row|^2, row = k + 8*hi (broadcast via shuffle) -
    v8f c;
#pragma unroll
    for (int k = 0; k < 8; ++k)
        c[k] = __shfl(rn2, k + (hi ? 8 : 0), 32);

    const int nM     = nmm[b];
    const int mBase0 = mc * MCHUNK;

    float s1[8], s3[8], tx[8], ty[8], tz[8];
#pragma unroll
    for (int k = 0; k < 8; ++k) { s1[k] = s3[k] = tx[k] = ty[k] = tz[k] = 0.0f; }

    for (int mb = mBase0; mb < mBase0 + MCHUNK; mb += 16) {
        const int jm = mb + col;
        const float* mp = posM + ((size_t)b * Mc + jm) * 3;
        const float mx = mp[0], my = mp[1], mz = mp[2];
        const float qj = (jm < nM) ? chg[(size_t)b * Mc + jm] : 0.0f;
        const float mn2 = mx * mx + my * my + mz * mz;

        // ---- B operand: col = lane&15, K = (-2x_j, -2y_j, -2z_j, |r_j|^2)
        v2f bb;
        bb.x = hi ? -2.0f * mz : -2.0f * mx;
        bb.y = hi ? mn2        : -2.0f * my;

        // d2[k] = |r_i|^2 + |r_j|^2 - 2 r_i.r_j   (row = k+8*hi, col = lane&15)
        v8f d2 = __builtin_amdgcn_wmma_f32_16x16x4_f32(
            /*neg_a=*/false, a, /*neg_b=*/false, bb,
            /*c_mod=*/(short)0, c, /*reuse_a=*/false, /*reuse_b=*/false);

#pragma unroll
        for (int k = 0; k < 8; ++k) {
            float dd   = fmaxf(d2[k], 0.0f);
            float dist = __builtin_amdgcn_sqrtf(dd) + EPSV;
            float inv  = __builtin_amdgcn_rcpf(dist);
            float w1   = qj * inv;              // q/d
            float w3   = w1 * inv * inv;        // q/d^3
            s1[k] += w1;
            s3[k] += w3;
            tx[k] += w3 * mx;
            ty[k] += w3 * my;
            tz[k] += w3 * mz;
        }
    }

    // ---- reduce the 16 columns within each 16-lane half --------------------
#pragma unroll
    for (int k = 0; k < 8; ++k) {
#pragma unroll
        for (int off = 8; off >= 1; off >>= 1) {
            s1[k] += __shfl_xor(s1[k], off, 32);
            s3[k] += __shfl_xor(s3[k], off, 32);
            tx[k] += __shfl_xor(tx[k], off, 32);
            ty[k] += __shfl_xor(ty[k], off, 32);
            tz[k] += __shfl_xor(tz[k], off, 32);
        }
    }

    // ---- lane 0 writes rows rowBase..+7, lane 16 writes rowBase+8..+15 -----
    if ((lane & 15) == 0) {
#pragma unroll
        for (int k = 0; k < 8; ++k) {
            const int row = rowBase + k + (hi ? 8 : 0);
            const float* qi = posQ + ((size_t)b * Nc + row) * 3;
            const float Ex = qi[0] * s3[k] - tx[k];
            const float Ey = qi[1] * s3[k] - ty[k];
            const float Ez = qi[2] * s3[k] - tz[k];

            atomicAdd(out + (size_t)b * Nc + row, s1[k]);
            float* ef = out + (size_t)Bc * Nc + ((size_t)b * Nc + row) * 3;
            atomicAdd(ef + 0, Ex);
            atomicAdd(ef + 1, Ey);
            atomicAdd(ef + 2, Ez);
        }
    }
}

extern "C" void kernel_launch(void* const* d_in, const int* in_sizes, int n_in,
                              void* d_out, int out_size, void* d_ws, size_t ws_size,
                              hipStream_t stream) {
    const float* posQ = (const float*)d_in[0];   // [32,128,3]
    const float* posM = (const float*)d_in[1];   // [32,4096,3]
    const float* chg  = (const float*)d_in[2];   // [32,4096]
    const int*   nmm  = (const int*)d_in[3];     // [32]
    float* out = (float*)d_out;                  // 4096 + 12288 floats

    // zero accumulation targets (harness poisons d_out with 0xAA)
    zero_kernel<<<(out_size + 255) / 256, 256, 0, stream>>>(out, out_size);

    const int totalWaves = Bc * ROW_TILES * MSPLIT;          // 2048 waves
    const int blocks     = totalWaves / WAVES_PER_BLOCK;     // 256 blocks
    mm_features_kernel<<<blocks, 32 * WAVES_PER_BLOCK, 0, stream>>>(
        posQ, posM, chg, nmm, out);
}